// GRAPHLayer_63513976373392
// MI455X (gfx1250) — compile-verified
//
#include <hip/hip_runtime.h>

typedef float v2f __attribute__((ext_vector_type(2)));
typedef float v8f __attribute__((ext_vector_type(8)));

#define DIM 256

// ---------------- degree / normalization ----------------
__global__ void deg_init_kernel(float* deg, int n) {
    int i = blockIdx.x * blockDim.x + threadIdx.x;
    if (i < n) deg[i] = 1.0f;  // self-loop contribution
}

__global__ void deg_count_kernel(const long long* __restrict__ dst, float* deg, int E) {
    int e = blockIdx.x * blockDim.x + threadIdx.x;
    if (e < E) unsafeAtomicAdd(&deg[dst[e]], 1.0f);
}

__global__ void deg_to_dinv_kernel(float* deg, int n) {
    int i = blockIdx.x * blockDim.x + threadIdx.x;
    if (i < n) deg[i] = rsqrtf(deg[i]);  // deg >= 1 always
}

// ---------------- fp32 WMMA GEMM: Y = X @ W (+bias, relu) ----------------
// X: [N,256] row-major, W: [256,256] row-major, Y: [N,256]
// grid = (N/16, 2), block = 256 threads (8 waves). Wave w computes the
// 16x16 output tile at rows [16*bx, +16), cols [128*by + 16*w, +16).
__global__ __launch_bounds__(256) void gemm_wmma_f32_kernel(
    const float* __restrict__ X, const float* __restrict__ W,
    const float* __restrict__ bias, float* __restrict__ Y, int fuse_bias_relu) {
    // 16x256 A tile; row stride 260 floats -> rows skew by 4 banks (64-bank LDS),
    // so the 16-lane column reads below are conflict-free; 260%4==0 keeps float4 alignment.
    __shared__ __align__(16) float As[16][260];

    const int tid  = threadIdx.x;
    const int row0 = blockIdx.x * 16;

    // Cooperative A-tile load: 16 rows x 64 float4 = 1024 float4, 4 per thread.
    for (int i = tid; i < 16 * 64; i += 256) {
        const int r = i >> 6;
        const int c = (i & 63) << 2;
        const float4 v = *(const float4*)(X + (size_t)(row0 + r) * DIM + c);
        *(float4*)(&As[r][c]) = v;
    }
    __syncthreads();

    const int wave = tid >> 5;
    const int lane = tid & 31;
    const int lm   = lane & 15;           // M for A-frag, N for B-frag/C
    const int kh   = (lane >> 4) << 1;    // K sub-offset: lanes 0-15 -> 0, lanes 16-31 -> 2
    const int col  = blockIdx.y * 128 + wave * 16 + lm;

    v8f acc = {0.f, 0.f, 0.f, 0.f, 0.f, 0.f, 0.f, 0.f};

    for (int k0 = 0; k0 < DIM; k0 += 4) {
        v2f a, b;
        // A 16x4 fragment (ISA 7.12.2): lane holds row lm, K = kh..kh+1
        a[0] = As[lm][k0 + kh];
        a[1] = As[lm][k0 + kh + 1];
        // B 4x16 fragment: lane holds col `col`, K = kh..kh+1 (row-striped across lanes)
        b[0] = W[(size_t)(k0 + kh) * DIM + col];
        b[1] = W[(size_t)(k0 + kh + 1) * DIM + col];
        acc = __builtin_amdgcn_wmma_f32_16x16x4_f32(
            /*neg_a=*/false, a, /*neg_b=*/false, b,
            /*c_mod=*/(short)0, acc, /*reuse_a=*/false, /*reuse_b=*/false);
    }

    // C/D layout: VGPR r -> row r (lanes 0-15) or r+8 (lanes 16-31), col = lane&15
    const int mhalf = (lane >> 4) << 3;
    const float bj = fuse_bias_relu ? bias[col] : 0.0f;
#pragma unroll
    for (int r = 0; r < 8; ++r) {
        float v = acc[r] + bj;
        if (fuse_bias_relu) v = fmaxf(v, 0.0f);
        Y[(size_t)(row0 + r + mhalf) * DIM + col] = v;
    }
}

// ---------------- edge scatter: agg[dst] += h[src] * dinv[src]*dinv[dst] ----------------
// One wave per edge; lane-strided coalesced f32 atomics (L2-resident accumulator).
__global__ __launch_bounds__(256) void scatter_edges_kernel(
    const long long* __restrict__ src, const long long* __restrict__ dst,
    const float* __restrict__ h, const float* __restrict__ dinv,
    float* __restrict__ agg, int E) {
    const int e    = (int)((blockIdx.x * 256u + threadIdx.x) >> 5);
    const int lane = threadIdx.x & 31;
    if (e >= E) return;
    const long long s = src[e];
    const long long d = dst[e];
    const float nm = dinv[s] * dinv[d];
    const float* __restrict__ hs = h + (size_t)s * DIM;
    float* ad = agg + (size_t)d * DIM;
#pragma unroll
    for (int j = lane; j < DIM; j += 32)
        unsafeAtomicAdd(&ad[j], hs[j] * nm);
}

// ---------------- combine: y = relu(agg + h * dinv^2 (self loop) + bias) ----------------
__global__ void combine_kernel(const float* __restrict__ agg, const float* __restrict__ h,
                               const float* __restrict__ dinv, const float* __restrict__ bias,
                               float* __restrict__ y, int total) {
    int idx = blockIdx.x * blockDim.x + threadIdx.x;
    if (idx >= total) return;
    const int i = idx >> 8;
    const int j = idx & 255;
    const float di = dinv[i];
    const float v = agg[idx] + h[idx] * di * di + bias[j];
    y[idx] = fmaxf(v, 0.0f);
}

extern "C" void kernel_launch(void* const* d_in, const int* in_sizes, int n_in,
                              void* d_out, int out_size, void* d_ws, size_t ws_size,
                              hipStream_t stream) {
    const float*      x_in = (const float*)d_in[0];       // [N,256] fp32
    const long long*  eidx = (const long long*)d_in[1];   // [2,E] int64
    const float*      W0   = (const float*)d_in[2];       // [256,256]
    const float*      b0   = (const float*)d_in[3];       // [256]
    const float*      Wc   = (const float*)d_in[4];       // [2,256,256]
    const float*      bc   = (const float*)d_in[5];       // [2,256]
    float*            out  = (float*)d_out;               // [N,256]

    const int N = in_sizes[0] / DIM;   // 50000 (divisible by 16)
    const int E = in_sizes[1] / 2;     // 800000
    const long long* src = eidx;
    const long long* dst = eidx + E;

    // Workspace layout: h [N*256] | agg [N*256] | dinv [N]
    float* h    = (float*)d_ws;
    float* agg  = h + (size_t)N * DIM;
    float* dinv = agg + (size_t)N * DIM;

    const dim3 gemm_grid(N / 16, 2);
    const int  total = N * DIM;

    // 1) Symmetric normalization coefficients (shared by both conv layers).
    deg_init_kernel<<<(N + 255) / 256, 256, 0, stream>>>(dinv, N);
    deg_count_kernel<<<(E + 255) / 256, 256, 0, stream>>>(dst, dinv, E);
    deg_to_dinv_kernel<<<(N + 255) / 256, 256, 0, stream>>>(dinv, N);

    // 2) Initial dense projection: x = relu(input @ weight + bias)  (x lives in d_out)
    gemm_wmma_f32_kernel<<<gemm_grid, 256, 0, stream>>>(x_in, W0, b0, out, 1);

    // 3) Two GCN conv layers.
    for (int l = 0; l < 2; ++l) {
        const float* Wl = Wc + (size_t)l * DIM * DIM;
        const float* bl = bc + (size_t)l * DIM;
        // h = x @ W_l   (bias deferred to combine)
        gemm_wmma_f32_kernel<<<gemm_grid, 256, 0, stream>>>(out, Wl, bl, h, 0);
        // agg = 0
        hipMemsetAsync(agg, 0, (size_t)total * sizeof(float), stream);
        // agg[dst] += h[src] * dinv[src]*dinv[dst]   (one wave per edge)
        scatter_edges_kernel<<<(E + 7) / 8, 256, 0, stream>>>(src, dst, h, dinv, agg, E);
        // x = relu(agg + h*dinv^2 + b_l)   (overwrites d_out; reads only h/agg/dinv)
        combine_kernel<<<(total + 255) / 256, 256, 0, stream>>>(agg, h, dinv, bl, out, total);
    }
}